// GFL_62895501082683
// MI455X (gfx1250) — compile-verified
//
#include <hip/hip_runtime.h>

#define N_NODES  50000
#define N_EDGES  1600000
#define F_IN     128
#define F_OUT    8
#define K_TAPS   4
#define MAX_P    25        // F_OUT*(K_TAPS-1)+1 highest power of S needed
#define NPAD     50016     // N_NODES + 16, a multiple of 16 (WMMA tiles)
#define KPAD     28        // K = 25 padded to multiple of 4 for 16x16x4 WMMA
#define NSLOTS   (KPAD + 1) // slot 0 = v, slots 1..25 = S^p v, 26..28 = zero pad

typedef __attribute__((ext_vector_type(2))) float v2f;
typedef __attribute__((ext_vector_type(8))) float v8f;

// ---------------------------------------------------------------------------
// Kernel 1: v[n] = sum_f x[n,f].  One wave32 per node: each lane loads one
// float4 (128 floats = 32 lanes * 4), then a 5-step butterfly reduction.
// ---------------------------------------------------------------------------
__global__ void gf_rowsum(const float* __restrict__ x, float* __restrict__ v)
{
    int wave = (blockIdx.x * blockDim.x + threadIdx.x) >> 5;   // node id
    int lane = threadIdx.x & 31;
    if (wave >= N_NODES) return;                               // wave-uniform
    const float4* row = (const float4*)(x + (size_t)wave * F_IN);
    float4 t = row[lane];                                      // global_load_b128
    float s = t.x + t.y + t.z + t.w;
#pragma unroll
    for (int off = 16; off > 0; off >>= 1)
        s += __shfl_xor(s, off, 32);
    if (lane == 0) v[wave] = s;
}

// ---------------------------------------------------------------------------
// Kernel 2: zero a region (SpMV scatter targets + zero-pad slots).
// ---------------------------------------------------------------------------
__global__ void gf_zero(float* __restrict__ p, int n)
{
    int i = blockIdx.x * blockDim.x + threadIdx.x;
    if (i < n) p[i] = 0.0f;
}

// ---------------------------------------------------------------------------
// Kernel 3: build the padded tap matrix, column-major:
//   Wt[o*KPAD + p] = coeff[o, p - 3o] if 0 <= p-3o < 4 and p < 25, o < 8; else 0
// 448 elements total; makes the combine kernel's B loads branch-free b64s.
// ---------------------------------------------------------------------------
__global__ void gf_build_w(const float* __restrict__ coeff, float* __restrict__ Wt)
{
    int i = blockIdx.x * blockDim.x + threadIdx.x;
    if (i >= 16 * KPAD) return;
    int o = i / KPAD;          // column (output feature), 0..15
    int p = i % KPAD;          // row (power index), 0..27
    int k = p - 3 * o;
    float val = 0.0f;
    if (o < F_OUT && p < MAX_P && k >= 0 && k < K_TAPS)
        val = coeff[o * K_TAPS + k];
    Wt[i] = val;
}

// ---------------------------------------------------------------------------
// Kernel 4: vector SpMV  vout[rows[e]] += vals[e] * vin[cols[e]].
// Edge arrays read coalesced; vin/vout (200 KB) are L2-resident; the scatter
// uses native global_atomic_add_f32 (L2 atomic units, ~32-way node reuse).
// ---------------------------------------------------------------------------
__global__ void gf_spmv(const int* __restrict__ rows, const int* __restrict__ cols,
                        const float* __restrict__ vals,
                        const float* __restrict__ vin, float* __restrict__ vout)
{
    int e = blockIdx.x * blockDim.x + threadIdx.x;
    if (e >= N_EDGES) return;
    float contrib = vals[e] * vin[cols[e]];
    unsafeAtomicAdd(&vout[rows[e]], contrib);
}

// ---------------------------------------------------------------------------
// Kernel 5: y(50000x8) = s^T(50000x25) @ W(25x8) via V_WMMA_F32_16X16X4_F32.
// One wave per 16-node tile, K padded to 28 -> 7 WMMA steps, branch-free body.
//   A layout: lane m = L&15 holds row m; L>=16 selects K-pair {2,3} vs {0,1}.
//   B layout: lane holds column m; L>=16 selects K-row pair {2,3} vs {0,1}.
//   D layout: VGPR j -> row (j + 8*(L>=16)), column m.
// ---------------------------------------------------------------------------
__global__ void gf_combine(const float* __restrict__ sbuf,   // slot p = S^p v, stride NPAD
                           const float* __restrict__ Wt,     // [16][KPAD] column-major taps
                           float* __restrict__ y)
{
    int wave = (blockIdx.x * blockDim.x + threadIdx.x) >> 5;  // tile id
    int lane = threadIdx.x & 31;
    const int nTiles = (N_NODES + 15) / 16;
    if (wave >= nTiles) return;                               // wave-uniform exit

    int hi = lane >> 4;        // 0: K {0,1}; 1: K {2,3} (and D rows +8)
    int m  = lane & 15;        // A row / B column / D column
    int n0 = wave * 16;

    const float* aBase = sbuf + n0 + m;          // + (p_idx+1)*NPAD per element
    const float* wCol  = Wt + m * KPAD;          // contiguous taps for column m

    v8f c = {};
#pragma unroll
    for (int k0 = 0; k0 < KPAD; k0 += 4) {
        int ka = k0 + 2 * hi;                    // even -> b64-aligned W pair
        v2f a, b;
        a.x = aBase[(size_t)(ka + 1) * NPAD];    // A[m][ka]   (slots <= 28, zero-padded)
        a.y = aBase[(size_t)(ka + 2) * NPAD];    // A[m][ka+1]
        v2f wpair = *(const v2f*)(wCol + ka);    // B[ka][m], B[ka+1][m] in one b64
        b.x = wpair.x;
        b.y = wpair.y;
        c = __builtin_amdgcn_wmma_f32_16x16x4_f32(
                /*neg_a=*/false, a, /*neg_b=*/false, b,
                /*c_mod=*/(short)0, c, /*reuse_a=*/false, /*reuse_b=*/false);
    }

    // Masked stores only after all WMMAs (EXEC divergence is legal here).
    if (m < F_OUT) {
#pragma unroll
        for (int j = 0; j < 8; ++j) {
            int node = n0 + j + 8 * hi;
            if (node < N_NODES) y[(size_t)node * F_OUT + m] = c[j];
        }
    }
}

// ---------------------------------------------------------------------------
extern "C" void kernel_launch(void* const* d_in, const int* in_sizes, int n_in,
                              void* d_out, int out_size, void* d_ws, size_t ws_size,
                              hipStream_t stream)
{
    const float* x     = (const float*)d_in[0];
    const int*   rows  = (const int*)d_in[1];
    const int*   cols  = (const int*)d_in[2];
    const float* vals  = (const float*)d_in[3];
    const float* coeff = (const float*)d_in[4];
    float*       y     = (float*)d_out;
    float*       sbuf  = (float*)d_ws;                     // NSLOTS * NPAD floats
    float*       Wt    = sbuf + (size_t)NSLOTS * NPAD;     // 16*KPAD floats

    // Zero the pad slots (26..28) so combine's tail loads read exact zeros.
    gf_zero<<<(3 * NPAD + 255) / 256, 256, 0, stream>>>(
        sbuf + (size_t)(MAX_P + 1) * NPAD, 3 * NPAD);

    // Dense padded tap matrix (column-major) for branch-free B loads.
    gf_build_w<<<2, 256, 0, stream>>>(coeff, Wt);

    // slot 0 = v = x.sum(axis=1)
    gf_rowsum<<<(N_NODES + 7) / 8, 256, 0, stream>>>(x, sbuf);

    // slots 1..25 = S^p v (serial chain; edge list + vectors are L2-resident)
    for (int p = 1; p <= MAX_P; ++p) {
        float*       outv = sbuf + (size_t)p * NPAD;
        const float* inv  = sbuf + (size_t)(p - 1) * NPAD;
        gf_zero<<<(NPAD + 255) / 256, 256, 0, stream>>>(outv, NPAD);
        gf_spmv<<<(N_EDGES + 255) / 256, 256, 0, stream>>>(rows, cols, vals, inv, outv);
    }

    // y = s^T @ W via WMMA
    const int nTiles = (N_NODES + 15) / 16;
    gf_combine<<<(nTiles * 32 + 255) / 256, 256, 0, stream>>>(sbuf, Wt, y);
}